// VolumetricRenderer_66348654788997
// MI455X (gfx1250) — compile-verified
//
#include <hip/hip_runtime.h>

// NeRF volumetric compositing: N=65536 rays, S=192 samples.
// Memory-bound streaming kernel (~302 MB @ 23.3 TB/s => ~13us floor).
// One wave32 per ray; 6 chunks of 32 samples; product scan via lane shuffles;
// density/dists staged to LDS via CDNA5 async global->LDS loads when available.

#define NERF_S 192
#define WAVES_PER_BLOCK 8
#define CHUNKS (NERF_S / 32)

// ---- CDNA5 async global->LDS availability probe -----------------------------
#if defined(__has_builtin)
#if __has_builtin(__builtin_amdgcn_global_load_async_to_lds_b128) && \
    __has_builtin(__builtin_amdgcn_global_load_async_to_lds_b64)
#define HAVE_ASYNC_LDS 1
#endif
#endif
#ifndef HAVE_ASYNC_LDS
#define HAVE_ASYNC_LDS 0
#endif

#if HAVE_ASYNC_LDS
// Builtin signatures (from hipcc diagnostic): b128 takes int-vector(4) pointers,
// b64 takes int-vector(2) pointers, src in AS1 (global), dst in AS3 (LDS),
// then two immediate ints (offset, cpol).
typedef int v4i_t __attribute__((vector_size(16)));
typedef int v2i_t __attribute__((vector_size(8)));
typedef __attribute__((address_space(1))) v4i_t g_v4i;
typedef __attribute__((address_space(3))) v4i_t l_v4i;
typedef __attribute__((address_space(1))) v2i_t g_v2i;
typedef __attribute__((address_space(3))) v2i_t l_v2i;
#endif

__device__ __forceinline__ void wait_asynccnt0() {
#if defined(__has_builtin)
#if __has_builtin(__builtin_amdgcn_s_wait_asynccnt)
    __builtin_amdgcn_s_wait_asynccnt(0);
#else
    asm volatile("s_wait_asynccnt 0" ::: "memory");
#endif
#else
    asm volatile("s_wait_asynccnt 0" ::: "memory");
#endif
}

__global__ __launch_bounds__(256, 4) void VolumetricRenderer_66348654788997_kernel(
    const float* __restrict__ rgb,      // (N, S, 3)
    const float* __restrict__ density,  // (N, S)
    const float* __restrict__ dists,    // (N, S)
    float* __restrict__ rgb_map,        // (N, 3)
    float* __restrict__ weights,        // (N, S)
    int n_rays) {
#if HAVE_ASYNC_LDS
    __shared__ float s_den[WAVES_PER_BLOCK][NERF_S];
    __shared__ float s_dst[WAVES_PER_BLOCK][NERF_S];
#endif

    const int lane = threadIdx.x & 31;
    const int wave = threadIdx.x >> 5;
    const int ray  = blockIdx.x * WAVES_PER_BLOCK + wave;
    if (ray >= n_rays) return;

    const size_t rbase = (size_t)ray * NERF_S;
    const float* gden = density + rbase;
    const float* gdst = dists + rbase;
    const float* grgb = rgb + rbase * 3;
    float* wout = weights + rbase;

#if HAVE_ASYNC_LDS
    {
        // Samples [0,128): b128 (4 floats/lane). Samples [128,192): b64 (2 floats/lane).
        __builtin_amdgcn_global_load_async_to_lds_b128(
            (g_v4i*)(gden + lane * 4), (l_v4i*)&s_den[wave][lane * 4], 0, 0);
        __builtin_amdgcn_global_load_async_to_lds_b64(
            (g_v2i*)(gden + 128 + lane * 2), (l_v2i*)&s_den[wave][128 + lane * 2], 0, 0);
        __builtin_amdgcn_global_load_async_to_lds_b128(
            (g_v4i*)(gdst + lane * 4), (l_v4i*)&s_dst[wave][lane * 4], 0, 0);
        __builtin_amdgcn_global_load_async_to_lds_b64(
            (g_v2i*)(gdst + 128 + lane * 2), (l_v2i*)&s_dst[wave][128 + lane * 2], 0, 0);
        wait_asynccnt0();  // same wave consumes its own LDS tile; no barrier needed
    }
#endif

    float accR = 0.0f, accG = 0.0f, accB = 0.0f, accW = 0.0f;
    float prefix = 1.0f;  // carried transmittance product across chunks

#pragma unroll
    for (int c = 0; c < CHUNKS; ++c) {
        const int idx = c * 32 + lane;

        // Prefetch next chunk of the (dominant) rgb stream past the serial scan dep.
        if (c + 1 < CHUNKS) {
            __builtin_prefetch(grgb + (size_t)(idx + 32) * 3, 0, 1);
        }

#if HAVE_ASYNC_LDS
        const float d = s_den[wave][idx];
        const float t = s_dst[wave][idx];
#else
        const float d = __builtin_nontemporal_load(gden + idx);
        const float t = __builtin_nontemporal_load(gdst + idx);
#endif

        // one_minus_alpha = exp(-softplus(d) * t), softplus computed stably.
        const float sp    = fmaxf(d, 0.0f) + __logf(1.0f + __expf(-fabsf(d)));
        const float oma   = __expf(-sp * t);
        const float alpha = 1.0f - oma;

        // Inclusive product scan across the wave (5 shuffle steps, wave32).
        float incl = oma;
#pragma unroll
        for (int off = 1; off < 32; off <<= 1) {
            const float v = __shfl_up(incl, off, 32);
            if (lane >= off) incl *= v;
        }
        float excl = __shfl_up(incl, 1, 32);
        if (lane == 0) excl = 1.0f;

        const float T = prefix * excl;   // exclusive transmittance
        const float w = alpha * T;
        __builtin_nontemporal_store(w, wout + idx);

        const float r0 = __builtin_nontemporal_load(grgb + (size_t)idx * 3 + 0);
        const float r1 = __builtin_nontemporal_load(grgb + (size_t)idx * 3 + 1);
        const float r2 = __builtin_nontemporal_load(grgb + (size_t)idx * 3 + 2);
        accR = fmaf(w, r0, accR);
        accG = fmaf(w, r1, accG);
        accB = fmaf(w, r2, accB);
        accW += w;

        prefix *= __shfl(incl, 31, 32);  // carry full-chunk product forward
    }

    // Butterfly reduction across the wave.
#pragma unroll
    for (int off = 16; off >= 1; off >>= 1) {
        accR += __shfl_xor(accR, off, 32);
        accG += __shfl_xor(accG, off, 32);
        accB += __shfl_xor(accB, off, 32);
        accW += __shfl_xor(accW, off, 32);
    }

    if (lane == 0) {
        const float bg = 1.0f - accW;  // white background compositing
        float* o = rgb_map + (size_t)ray * 3;
        o[0] = accR + bg;
        o[1] = accG + bg;
        o[2] = accB + bg;
    }
}

extern "C" void kernel_launch(void* const* d_in, const int* in_sizes, int n_in,
                              void* d_out, int out_size, void* d_ws, size_t ws_size,
                              hipStream_t stream) {
    (void)n_in; (void)out_size; (void)d_ws; (void)ws_size;
    const float* rgb     = (const float*)d_in[0];  // (N, S, 3)
    const float* density = (const float*)d_in[1];  // (N, S, 1)
    const float* dists   = (const float*)d_in[2];  // (N, S)

    const int n_rays = in_sizes[2] / NERF_S;       // dists has N*S elements

    float* out     = (float*)d_out;
    float* rgb_map = out;                          // first output: (N, 3)
    float* weights = out + (size_t)n_rays * 3;     // second output: (N, S)

    dim3 block(256);  // 8 wave32's -> 8 rays per block
    dim3 grid((n_rays + WAVES_PER_BLOCK - 1) / WAVES_PER_BLOCK);
    VolumetricRenderer_66348654788997_kernel<<<grid, block, 0, stream>>>(
        rgb, density, dists, rgb_map, weights, n_rays);
}